// FrequencyEnergyIndicator_10763188044295
// MI455X (gfx1250) — compile-verified
//
#include <hip/hip_runtime.h>
#include <hip/hip_bf16.h>

// ---------------------------------------------------------------------------
// FrequencyEnergyIndicator for MI455X (gfx1250, wave32, WMMA).
//
// Separable Gaussian blurs (sigma 2/4/8, k 17/33/65, zero padding) expressed
// as banded-Toeplitz 16x16 tile matmuls on V_WMMA_F32_16X16X4_F32 (exact fp32).
// Phase 1: horizontal blur  z -> H1,H2,H3   (d_ws, 192 MiB)
// Phase 2: vertical blur    H -> p1,p2,p3, bands [p3, p2-p3, p1-p2] -> d_out,
//          per-block energy partials -> d_ws (deterministic tree reduction)
// Phase 3: deterministic batch reduction + normalization -> e_t (d_out[0:96])
//
// All per-sigma constants are constexpr + template parameters so the chunk
// loop fully unrolls; Toeplitz out-of-band taps come from zero guard bands in
// LDS (unconditional ds_load_b32, no exec branching -> EXEC stays all-ones
// around every WMMA).
// ---------------------------------------------------------------------------

typedef __attribute__((ext_vector_type(2))) float v2f;
typedef __attribute__((ext_vector_type(8))) float v8f;

#define IMG_H      256
#define IMG_W      256
#define NIMG       256                 // B*C = 32*8
#define IMG_ELEMS  65536               // 256*256
#define SIG_ELEMS  (NIMG * IMG_ELEMS)  // one blurred tensor: 16,777,216 floats
#define BAND_ELEMS SIG_ELEMS
#define MAXPAD     32
#define NCHUNK     20                  // (16 + 2*MAXPAD)/4 contraction chunks
#define NBLOCKS    4096                // 256 images * 16 strips
#define ET_OFF     96                  // e_t floats at front of d_out

// Per-sigma compile-time constants (sigma = 2,4,8; k = 17,33,65).
constexpr int PADS[3] = {8, 16, 32};
constexpr int MLO[3]  = {6, 4, 0};     // first active K-chunk
constexpr int MHI[3]  = {13, 15, 19};  // last active K-chunk
// LDS weight table with zero guard bands: each region = 16 zeros | k taps |
// 17 zeros. WBASE[s] points at tap index 0; indices in [-15, 2p+16] are safe.
constexpr int WBASE[3] = {16, 66, 132};
constexpr int LDSW     = 214;          // 50 + 66 + 98

__device__ __forceinline__ v8f wmma_f32x4(v2f a, v2f b, v8f c) {
  // V_WMMA_F32_16X16X4_F32: D(16x16,f32) = A(16x4,f32) x B(4x16,f32) + C
  return __builtin_amdgcn_wmma_f32_16x16x4_f32(
      /*neg_a=*/false, a, /*neg_b=*/false, b,
      /*c_mod=*/(short)0, c, /*reuse_a=*/false, /*reuse_b=*/false);
}

// Build normalized 1D Gaussian taps (with zero guards) in LDS.
__device__ __forceinline__ void build_weights(float* wts, int tid) {
  if (tid < LDSW) wts[tid] = 0.0f;     // zero guards (and everything else)
  __syncthreads();
  if (tid < 115) {
    int s, i, p;
    if (tid < 17)      { s = 0; i = tid;       p = 8;  }
    else if (tid < 50) { s = 1; i = tid - 17;  p = 16; }
    else               { s = 2; i = tid - 50;  p = 32; }
    float sigma = (float)(2 << s);     // 2, 4, 8
    float x = (float)(i - p);
    wts[WBASE[s] + i] = expf(-(x * x) / (2.0f * sigma * sigma));
  }
  __syncthreads();
  __shared__ float s_inv[3];
  if (tid < 3) {
    int k = 2 * PADS[tid] + 1;
    float s = 0.0f;
    for (int j = 0; j < k; ++j) s += wts[WBASE[tid] + j];
    s_inv[tid] = 1.0f / s;
  }
  __syncthreads();
  if (tid < 115) {
    int s = (tid < 17) ? 0 : (tid < 50 ? 1 : 2);
    int i = (s == 0) ? tid : (s == 1 ? tid - 17 : tid - 50);
    wts[WBASE[s] + i] *= s_inv[s];
  }
  __syncthreads();
}

// One 16x16x4 WMMA step of the horizontal pass for sigma index S.
// B = Toeplitz taps from guarded LDS (unconditional loads).
template <int S>
__device__ __forceinline__ void hz_step(int m, int t0, v2f a, const float* wts,
                                        v8f& acc) {
  if (m < MLO[S] || m > MHI[S]) return;    // folds after full unroll
  const int i0 = t0 + PADS[S];             // in [-15, 2p+15]; guards cover it
  v2f b;
  b.x = wts[WBASE[S] + i0];
  b.y = wts[WBASE[S] + i0 + 1];
  acc = wmma_f32x4(a, b, acc);
}

// One 16x16x4 WMMA step of the vertical pass for sigma index S.
// A = Toeplitz taps from guarded LDS; B = two rows of H_S (clamped + select).
template <int S>
__device__ __forceinline__ void vt_step(int m, int j, int tA0, int col,
                                        const float* __restrict__ Hs,
                                        const float* wts, v8f& acc) {
  if (m < MLO[S] || m > MHI[S]) return;
  const int i0 = tA0 + PADS[S];
  v2f a;
  a.x = wts[WBASE[S] + i0];
  a.y = wts[WBASE[S] + i0 + 1];
  const int jc0 = min(max(j, 0), IMG_H - 1);
  const int jc1 = min(max(j + 1, 0), IMG_H - 1);
  float bx = Hs[(size_t)jc0 * IMG_W + col];
  float by = Hs[(size_t)jc1 * IMG_W + col];
  v2f b;
  b.x = ((unsigned)j       < (unsigned)IMG_H) ? bx : 0.0f;   // v_cndmask
  b.y = ((unsigned)(j + 1) < (unsigned)IMG_H) ? by : 0.0f;
  acc = wmma_f32x4(a, b, acc);
}

// ---------------------------------------------------------------------------
// Phase 1: horizontal blur.  One wave = one 16x16 output tile.
// Block = 512 threads = 16 waves = one 16-row strip of one image.
// ---------------------------------------------------------------------------
__global__ void __launch_bounds__(512)
fei_hblur_kernel(const float* __restrict__ z, float* __restrict__ hbuf) {
  __shared__ float wts[LDSW];
  const int tid = threadIdx.x;
  build_weights(wts, tid);

  const int img  = blockIdx.x >> 4;
  const int r0   = (blockIdx.x & 15) << 4;
  const int lane = tid & 31;
  const int c0   = (tid >> 5) << 4;    // wave id * 16
  const int half = lane >> 4;          // 0: K=0,1   1: K=2,3
  const int l16  = lane & 15;

  const float* srow = z + (size_t)img * IMG_ELEMS + (size_t)(r0 + l16) * IMG_W;

  v8f acc0 = (v8f)0.0f, acc1 = (v8f)0.0f, acc2 = (v8f)0.0f;

#pragma unroll
  for (int m = 0; m < NCHUNK; ++m) {
    const int relj = -MAXPAD + 4 * m + 2 * half;   // window col rel. to c0
    const int col  = c0 + relj;                    // even
    const int colc = min(max(col, 0), IMG_W - 2);  // clamped, in-range
    v2f a = *(const v2f*)(srow + colc);            // unconditional b64 load
    const bool av = (unsigned)col < (unsigned)IMG_W;
    a.x = av ? a.x : 0.0f;                         // v_cndmask, no exec change
    a.y = av ? a.y : 0.0f;
    const int t0 = relj - l16;                     // (j - c) for this lane
    hz_step<0>(m, t0, a, wts, acc0);
    hz_step<1>(m, t0, a, wts, acc1);
    hz_step<2>(m, t0, a, wts, acc2);
  }

  // Store three horizontally-blurred tiles (C/D layout: VGPR i -> rows i, i+8)
  const size_t base = (size_t)img * IMG_ELEMS + (size_t)(c0 + l16);
#pragma unroll
  for (int i = 0; i < 8; ++i) {
    const size_t off = base + (size_t)(r0 + i + 8 * half) * IMG_W;
    hbuf[0 * (size_t)SIG_ELEMS + off] = acc0[i];
    hbuf[1 * (size_t)SIG_ELEMS + off] = acc1[i];
    hbuf[2 * (size_t)SIG_ELEMS + off] = acc2[i];
  }
}

// ---------------------------------------------------------------------------
// Phase 2: vertical blur + band formation + per-block energy partials.
// ---------------------------------------------------------------------------
__global__ void __launch_bounds__(512)
fei_vblur_kernel(const float* __restrict__ hbuf, float* __restrict__ out,
                 float* __restrict__ partials) {
  __shared__ float wts[LDSW];
  __shared__ float red[512];
  const int tid = threadIdx.x;
  build_weights(wts, tid);

  const int img  = blockIdx.x >> 4;
  const int r0   = (blockIdx.x & 15) << 4;
  const int lane = tid & 31;
  const int c0   = (tid >> 5) << 4;
  const int half = lane >> 4;
  const int l16  = lane & 15;
  const int col  = c0 + l16;

  const float* H0 = hbuf + 0 * (size_t)SIG_ELEMS + (size_t)img * IMG_ELEMS;
  const float* H1 = hbuf + 1 * (size_t)SIG_ELEMS + (size_t)img * IMG_ELEMS;
  const float* H2 = hbuf + 2 * (size_t)SIG_ELEMS + (size_t)img * IMG_ELEMS;

  v8f acc0 = (v8f)0.0f, acc1 = (v8f)0.0f, acc2 = (v8f)0.0f;

#pragma unroll
  for (int m = 0; m < NCHUNK; ++m) {
    const int relj = -MAXPAD + 4 * m + 2 * half;
    const int j    = r0 + relj;                    // input row for b.x (even)
    const int tA0  = relj - l16;                   // (j - r) for out row l16
    vt_step<0>(m, j, tA0, col, H0, wts, acc0);
    vt_step<1>(m, j, tA0, col, H1, wts, acc1);
    vt_step<2>(m, j, tA0, col, H2, wts, acc2);
  }

  // bands: [p3, p2-p3, p1-p2]
  const v8f b0 = acc2;
  const v8f b1 = acc1 - acc2;
  const v8f b2 = acc0 - acc1;

  float e0 = 0.0f, e1 = 0.0f, e2 = 0.0f;
  float* ob = out + ET_OFF;
  const size_t base = (size_t)img * IMG_ELEMS + (size_t)col;
#pragma unroll
  for (int i = 0; i < 8; ++i) {
    const size_t off = base + (size_t)(r0 + i + 8 * half) * IMG_W;
    ob[0 * (size_t)BAND_ELEMS + off] = b0[i];  e0 += b0[i] * b0[i];
    ob[1 * (size_t)BAND_ELEMS + off] = b1[i];  e1 += b1[i] * b1[i];
    ob[2 * (size_t)BAND_ELEMS + off] = b2[i];  e2 += b2[i] * b2[i];
  }

  // Deterministic per-block tree reduction of the three band energies.
  float ev[3] = {e0, e1, e2};
  for (int band = 0; band < 3; ++band) {
    red[tid] = ev[band];
    __syncthreads();
    for (int stp = 256; stp > 0; stp >>= 1) {
      if (tid < stp) red[tid] += red[tid + stp];
      __syncthreads();
    }
    if (tid == 0) partials[(size_t)blockIdx.x * 3 + band] = red[0];
    __syncthreads();
  }
}

// ---------------------------------------------------------------------------
// Phase 3: per-batch reduction (128 blocks/batch, fixed order) + normalize.
// ---------------------------------------------------------------------------
__global__ void fei_finalize_kernel(const float* __restrict__ partials,
                                    float* __restrict__ out) {
  const int b = threadIdx.x;          // batch index 0..31
  if (b >= 32) return;
  float e[3] = {0.0f, 0.0f, 0.0f};
  const int blk0 = b * 128;           // 8 channels * 16 strips per batch
  for (int blk = blk0; blk < blk0 + 128; ++blk)
    for (int s = 0; s < 3; ++s) e[s] += partials[(size_t)blk * 3 + s];
  const float tot = e[0] + e[1] + e[2] + 1e-8f;
  for (int s = 0; s < 3; ++s) out[b * 3 + s] = e[s] / tot;
}

extern "C" void kernel_launch(void* const* d_in, const int* in_sizes, int n_in,
                              void* d_out, int out_size, void* d_ws, size_t ws_size,
                              hipStream_t stream) {
  (void)in_sizes; (void)n_in; (void)out_size; (void)ws_size;
  const float* z = (const float*)d_in[0];
  float* out = (float*)d_out;
  float* ws  = (float*)d_ws;
  // ws layout: H1,H2,H3 (3 * 16,777,216 floats = 192 MiB) then 12,288 partials
  float* partials = ws + 3ull * SIG_ELEMS;

  fei_hblur_kernel<<<NBLOCKS, 512, 0, stream>>>(z, ws);
  fei_vblur_kernel<<<NBLOCKS, 512, 0, stream>>>(ws, out, partials);
  fei_finalize_kernel<<<1, 32, 0, stream>>>(partials, out);
}